// TestLFFD_88802743812249
// MI455X (gfx1250) — compile-verified
//
#include <hip/hip_runtime.h>
#include <math.h>
#include <stdint.h>

// LFFD post-processing, fused for MI455X (gfx1250, wave32).
// Kernel A: per-anchor softmax-max score -> order-preserving u32 key (8.4 MB ws).
//           Branchless branch-id (v_cmp sums) + cndmask parameter selects.
// Kernel B: one 32-wave workgroup per image; async-DMA the 262KB key row into
//           LDS (ASYNCcnt path), incremental 100-way argmax extraction using
//           wave32 shfl_xor butterflies (3 barriers/extraction), sparse decode
//           of the 100 winners, greedy NMS, outputs.

#define NBR    8
#define NANCH  65648
#define MAXB   100
#define BATCHN 32

struct Ptrs {
    const float* cls[NBR];
    const float* reg[NBR];
};

__device__ __forceinline__ uint32_t key_encode(float s) {
    uint32_t u = __float_as_uint(s);
    return (u & 0x80000000u) ? ~u : (u | 0x80000000u);
}
__device__ __forceinline__ float key_decode(uint32_t k) {
    uint32_t u = (k & 0x80000000u) ? (k & 0x7FFFFFFFu) : ~k;
    return __uint_as_float(u);
}

// Branchless branch id: sum of compares against compile-time offsets.
__device__ __forceinline__ int branch_of(int j) {
    int b = 0;
    b += (j >= 25281); b += (j >= 50562); b += (j >= 56803); b += (j >= 63044);
    b += (j >= 64565); b += (j >= 64926); b += (j >= 65287);
    return b;
}

template <typename T>
__device__ __forceinline__ T sel8(int b, T v0, T v1, T v2, T v3, T v4, T v5, T v6, T v7) {
    T r = v0;
    r = (b == 1) ? v1 : r;  r = (b == 2) ? v2 : r;  r = (b == 3) ? v3 : r;
    r = (b == 4) ? v4 : r;  r = (b == 5) ? v5 : r;  r = (b == 6) ? v6 : r;
    r = (b == 7) ? v7 : r;
    return r;
}

__device__ __forceinline__ int   off_of(int b) { return sel8(b, 0, 25281, 50562, 56803, 63044, 64565, 64926, 65287); }
__device__ __forceinline__ int   hw_of(int b)  { return sel8(b, 25281, 25281, 6241, 6241, 1521, 361, 361, 361); }
__device__ __forceinline__ int   w_of(int b)   { return sel8(b, 159, 159, 79, 79, 39, 19, 19, 19); }
__device__ __forceinline__ float st_of(int b)  { return sel8(b, 4.f, 4.f, 8.f, 8.f, 16.f, 32.f, 32.f, 32.f); }
__device__ __forceinline__ float rf2_of(int b) { return sel8(b, 27.5f, 35.5f, 55.5f, 71.5f, 111.5f, 191.5f, 255.5f, 319.5f); }

// Wave32 butterfly argmax on (key, idx), tie -> lower idx. No barriers.
__device__ __forceinline__ void wave_argmax(uint32_t& k, uint32_t& id) {
    #pragma unroll
    for (int m = 16; m > 0; m >>= 1) {
        uint32_t k2 = __shfl_xor(k, m, 32);
        uint32_t i2 = __shfl_xor(id, m, 32);
        if (k2 > k || (k2 == k && i2 < id)) { k = k2; id = i2; }
    }
}

// ---------------------------------------------------------------------------
// Kernel A: masked score -> sortable key for every (image, anchor).
// ---------------------------------------------------------------------------
__global__ void lffd_score_kernel(Ptrs P, uint32_t* __restrict__ keys) {
    int j   = blockIdx.x * blockDim.x + threadIdx.x;
    int img = blockIdx.y;
    if (j >= NANCH) return;

    int b  = branch_of(j);
    int p  = j - off_of(b);
    int hw = hw_of(b);
    const float* cb = sel8(b, P.cls[0], P.cls[1], P.cls[2], P.cls[3],
                              P.cls[4], P.cls[5], P.cls[6], P.cls[7]);
    const float* c  = cb + (size_t)img * 3 * hw + p;
    float l0 = c[0], l1 = c[hw], l2 = c[2 * hw];

    int   cl = 0;
    float m  = l0;
    if (l1 > m) { m = l1; cl = 1; }
    if (l2 > m) { m = l2; cl = 2; }
    // max softmax prob = 1 / sum(exp(li - lmax)); argmax(softmax) == argmax(logits)
    float s = 1.0f / (expf(l0 - m) + expf(l1 - m) + expf(l2 - m));
    if (cl == 0) s = -1.0f;   // background masked (reference: where(fg, s, -1))

    keys[(size_t)img * NANCH + j] = key_encode(s);
}

// ---------------------------------------------------------------------------
// Kernel B: per-image LDS-resident top-100 + sparse decode + NMS + outputs.
// Launched <<<32, 1024, NANCH*4>>>.
// ---------------------------------------------------------------------------
__launch_bounds__(1024, 1)
__global__ void lffd_topk_nms_kernel(Ptrs P, const uint32_t* __restrict__ keys,
                                     float* __restrict__ out) {
    extern __shared__ uint32_t skey[];           // NANCH keys (262,592 B)
    __shared__ uint32_t rkey[32];
    __shared__ uint32_t ridx[32];
    __shared__ uint32_t topIdx[MAXB];
    __shared__ uint32_t topKey[MAXB];
    __shared__ float bx1[MAXB], by1[MAXB], bx2[MAXB], by2[MAXB], ssc[MAXB];
    __shared__ int   scl[MAXB], skeep[MAXB];

    const int t    = threadIdx.x;
    const int lane = t & 31;
    const int wv   = t >> 5;
    const int img  = blockIdx.x;
    const uint32_t* krow = keys + (size_t)img * NANCH;

    // ---- Stage key row into LDS with CDNA5 async global->LDS DMA (b128). ----
    uint32_t lds_base = (uint32_t)(size_t)(void*)&skey[0];
    for (int k = t; k < (NANCH / 4); k += 1024) {          // 16412 x 16B = 262,592 B
        uint32_t    la = lds_base + (uint32_t)k * 16u;
        const void* ga = (const void*)(krow + (size_t)k * 4);
        asm volatile("global_load_async_to_lds_b128 %0, %1, off"
                     :: "v"(la), "v"(ga) : "memory");
    }
    asm volatile("s_wait_asynccnt 0" ::: "memory");
    __syncthreads();

    // ---- Per-thread cached chunk maximum (chunk = {t, t+1024, ...}). ----
    uint32_t bk = 0u, bj = (uint32_t)t;
    for (int j = t; j < NANCH; j += 1024) {
        uint32_t k = skey[j];
        if (k > bk) { bk = k; bj = (uint32_t)j; }          // strict > keeps lowest idx
    }

    // ---- 100 argmax extractions; shfl butterflies + one 32-entry LDS reduce. ----
    for (int i = 0; i < MAXB; ++i) {
        uint32_t k = bk, id = bj;
        wave_argmax(k, id);                                 // intra-wave, no barrier
        if (lane == 0) { rkey[wv] = k; ridx[wv] = id; }
        __syncthreads();
        if (wv == 0) {
            uint32_t k3 = rkey[lane], i3 = ridx[lane];
            wave_argmax(k3, i3);
            if (lane == 0) { rkey[0] = k3; ridx[0] = i3; }
        }
        __syncthreads();
        uint32_t wkey = rkey[0], widx = ridx[0];
        if (t == 0) { topIdx[i] = widx; topKey[i] = wkey; }
        if (t == (int)(widx & 1023u)) {                    // owner: remove + rescan
            skey[widx] = 0u;
            bk = 0u; bj = (uint32_t)t;
            for (int j = t; j < NANCH; j += 1024) {
                uint32_t kk = skey[j];
                if (kk > bk) { bk = kk; bj = (uint32_t)j; }
            }
        }
        __syncthreads();
    }

    // ---- Sparse decode of the 100 winners (threads 0..99). ----
    float tx1 = 0.f, ty1 = 0.f, tx2 = 0.f, ty2 = 0.f, tarea = 0.f;
    if (t < MAXB) {
        int   idx   = (int)topIdx[t];
        float score = key_decode(topKey[t]);

        int   b  = branch_of(idx);
        int   p  = idx - off_of(b);
        int   hw = hw_of(b);
        int   w  = w_of(b);
        float st = st_of(b);
        float cx = (float)(p % w) * st + 0.5f * st;
        float cy = (float)(p / w) * st + 0.5f * st;

        const float* cb = sel8(b, P.cls[0], P.cls[1], P.cls[2], P.cls[3],
                                  P.cls[4], P.cls[5], P.cls[6], P.cls[7]);
        const float* c  = cb + (size_t)img * 3 * hw + p;
        float l0 = c[0], l1 = c[hw], l2 = c[2 * hw];
        int   cl = 0;
        float m  = l0;
        if (l1 > m) { m = l1; cl = 1; }
        if (l2 > m) { m = l2; cl = 2; }

        const float* rb = sel8(b, P.reg[0], P.reg[1], P.reg[2], P.reg[3],
                                  P.reg[4], P.reg[5], P.reg[6], P.reg[7]);
        const float* r  = rb + (size_t)img * 4 * hw + p;
        float rf = rf2_of(b);
        tx1 = cx - r[0]      * rf;
        ty1 = cy - r[hw]     * rf;
        tx2 = cx + r[2 * hw] * rf;
        ty2 = cy + r[3 * hw] * rf;
        tarea = (tx2 - tx1 + 1.0f) * (ty2 - ty1 + 1.0f);

        bx1[t] = tx1; by1[t] = ty1; bx2[t] = tx2; by2[t] = ty2;
        ssc[t] = score; scl[t] = cl;
        skeep[t] = (score >= 0.35f && cl > 0) ? 1 : 0;
    }
    __syncthreads();

    // ---- Greedy NMS, exactly the reference suppression rule. ----
    for (int i = 0; i < MAXB - 1; ++i) {
        if (t < MAXB && t > i && skeep[i] && skeep[t]) {
            float xmin = fmaxf(bx1[i], tx1);
            float ymin = fmaxf(by1[i], ty1);
            float xmax = fminf(bx2[i], tx2);
            float ymax = fminf(by2[i], ty2);
            float inter = fmaxf(xmax - xmin, 0.0f) * fmaxf(ymax - ymin, 0.0f);
            float ai = (bx2[i] - bx1[i] + 1.0f) * (by2[i] - by1[i] + 1.0f);
            float iou = inter / (ai + tarea - inter);
            if (iou > 0.5f) skeep[t] = 0;
        }
        __syncthreads();
    }

    // ---- Outputs: scores | classes | boxes | keep, all as float. ----
    if (t < MAXB) {
        int   kp = skeep[t];
        int   o  = img * MAXB + t;
        float* oS = out;
        float* oC = out + BATCHN * MAXB;
        float* oB = out + 2 * BATCHN * MAXB;
        float* oK = out + 2 * BATCHN * MAXB + BATCHN * MAXB * 4;
        oS[o] = kp ? ssc[t] : 0.0f;
        oC[o] = kp ? (float)scl[t] : 0.0f;
        float* ob = oB + (size_t)o * 4;
        ob[0] = kp ? tx1 : 0.0f;
        ob[1] = kp ? ty1 : 0.0f;
        ob[2] = kp ? tx2 : 0.0f;
        ob[3] = kp ? ty2 : 0.0f;
        oK[o] = kp ? 1.0f : 0.0f;
    }
}

// ---------------------------------------------------------------------------
extern "C" void kernel_launch(void* const* d_in, const int* in_sizes, int n_in,
                              void* d_out, int out_size, void* d_ws, size_t ws_size,
                              hipStream_t stream) {
    (void)in_sizes; (void)n_in; (void)out_size; (void)ws_size;

    Ptrs P;
    for (int i = 0; i < NBR; ++i) {
        P.cls[i] = (const float*)d_in[i];
        P.reg[i] = (const float*)d_in[8 + i];
    }
    uint32_t* keys = (uint32_t*)d_ws;   // 32 * 65648 * 4 B = 8.4 MB

    dim3 gA((NANCH + 255) / 256, BATCHN);
    lffd_score_kernel<<<gA, 256, 0, stream>>>(P, keys);

    size_t shmem = (size_t)NANCH * sizeof(uint32_t);   // 262,592 B (< 320 KB LDS)
    lffd_topk_nms_kernel<<<BATCHN, 1024, shmem, stream>>>(P, keys, (float*)d_out);
}